// Attention_Decoder_72258529788593
// MI455X (gfx1250) — compile-verified
//
#include <hip/hip_runtime.h>
#include <math.h>

#define B_ 64
#define T_ 64
#define L_ 400
#define E_ 256
#define H_ 512

typedef __attribute__((ext_vector_type(2))) float v2f;
typedef __attribute__((ext_vector_type(8))) float v8f;

// ---------------------------------------------------------------------------
// Generic fp32 WMMA GEMM:  Out[M,N] = A[M,K] @ W[N,K]^T (+ A2 @ W2^T) (+ bias)
// grid = (N/16, M/64), block = 128 (4 waves, one 16x16 tile each).
// Uses V_WMMA_F32_16X16X4_F32 (native fp32 matrix op on CDNA5, wave32).
// A-frag layout: lanes 0-15 hold K={k,k+1}, lanes 16-31 hold K={k+2,k+3}.
// ---------------------------------------------------------------------------
__global__ __launch_bounds__(128) void wmma_gemm_f32(
    const float* __restrict__ A, int lda,
    const float* __restrict__ W, int ldw, int K,
    const float* __restrict__ A2, int lda2,
    const float* __restrict__ W2, int ldw2, int K2,
    const float* __restrict__ bias,
    float* __restrict__ Out, int ldo)
{
  const int wave = threadIdx.x >> 5;
  const int lane = threadIdx.x & 31;
  const int half = lane >> 4;        // 0: K pair {0,1}, 1: K pair {2,3}
  const int idx  = lane & 15;        // row (A) / col (B) within tile
  const int row0 = blockIdx.y * 64 + wave * 16;
  const int col0 = blockIdx.x * 16;

  v8f acc = {};
  {
    const float* ap = A + (size_t)(row0 + idx) * lda + 2 * half;
    const float* wp = W + (size_t)(col0 + idx) * ldw + 2 * half;
#pragma unroll 4
    for (int k = 0; k < K; k += 4) {
      v2f a = *(const v2f*)(ap + k);
      v2f b = *(const v2f*)(wp + k);
      acc = __builtin_amdgcn_wmma_f32_16x16x4_f32(
          false, a, false, b, (short)0, acc, false, false);
    }
  }
  if (A2) {
    const float* ap = A2 + (size_t)(row0 + idx) * lda2 + 2 * half;
    const float* wp = W2 + (size_t)(col0 + idx) * ldw2 + 2 * half;
#pragma unroll 4
    for (int k = 0; k < K2; k += 4) {
      v2f a = *(const v2f*)(ap + k);
      v2f b = *(const v2f*)(wp + k);
      acc = __builtin_amdgcn_wmma_f32_16x16x4_f32(
          false, a, false, b, (short)0, acc, false, false);
    }
  }
  // C/D layout: VGPR r -> M = r (lanes 0-15) or M = r+8 (lanes 16-31), N = idx
  const int col = col0 + idx;
  const float bv = bias ? bias[col] : 0.0f;
#pragma unroll
  for (int r = 0; r < 8; ++r) {
    const int row = row0 + r + 8 * half;
    Out[(size_t)row * ldo + col] = acc[r] + bv;
  }
}

// ---------------------------------------------------------------------------
// LSTM cell elementwise: gates -> h_new,c_new (+ concat buffers for dec/out)
// ---------------------------------------------------------------------------
__device__ __forceinline__ float sigm(float x) { return 1.0f / (1.0f + __expf(-x)); }

__global__ __launch_bounds__(256) void lstm_cell(
    const float* __restrict__ gates,   // [B, 4H] (no biases yet)
    const float* __restrict__ b_ih, const float* __restrict__ b_hh,
    float* __restrict__ h, float* __restrict__ c,
    float* __restrict__ hc,            // [B, 2H] = [h_new, c_new]
    float* __restrict__ hctx)          // [B, 3H], h part written here
{
  const int t = blockIdx.x * blockDim.x + threadIdx.x;  // 0..B*H-1
  const int b = t / H_, j = t % H_;
  const float* g = gates + (size_t)b * 4 * H_;
  const float gi = g[j]          + b_ih[j]          + b_hh[j];
  const float gf = g[H_ + j]     + b_ih[H_ + j]     + b_hh[H_ + j];
  const float gg = g[2 * H_ + j] + b_ih[2 * H_ + j] + b_hh[2 * H_ + j];
  const float go = g[3 * H_ + j] + b_ih[3 * H_ + j] + b_hh[3 * H_ + j];
  const float cn = sigm(gf) * c[t] + sigm(gi) * tanhf(gg);
  const float hn = sigm(go) * tanhf(cn);
  h[t] = hn; c[t] = cn;
  hc[(size_t)b * 2 * H_ + j] = hn;
  hc[(size_t)b * 2 * H_ + H_ + j] = cn;
  hctx[(size_t)b * 3 * H_ + j] = hn;
}

// ---------------------------------------------------------------------------
// Fused Bahdanau attention: one 1024-thread block (32 waves, fills a WGP)
// per batch row. enc_feat + encoder_states stay L2-resident (157MB < 192MB).
// ---------------------------------------------------------------------------
__global__ __launch_bounds__(1024) void attn_fused(
    const float* __restrict__ enc_feat,   // [B, L, H]
    const float* __restrict__ dec,        // [B, H]
    const float* __restrict__ v,          // [H]
    const float* __restrict__ mask,       // [B, L]
    const float* __restrict__ enc_states, // [B, L, 2H]
    float* __restrict__ attn_out,         // [B, L] slice of d_out at step t
    float* __restrict__ hctx)             // ctx written at [B, 3H] offset H
{
  __shared__ float s_e[L_];
  __shared__ float s_red[32];
  __shared__ float s_dec[H_];
  __shared__ float s_v[H_];
  __shared__ float s_bcast;

  const int b = blockIdx.x;
  const int tid = threadIdx.x;
  const int lane = tid & 31, wave = tid >> 5;

  if (tid < H_) { s_dec[tid] = dec[(size_t)b * H_ + tid]; s_v[tid] = v[tid]; }
  __syncthreads();

  // ---- scores e[l] = sum_h v[h] * tanh(enc_feat[b,l,h] + dec[b,h]) ----
  for (int l = wave; l < L_; l += 32) {
    const float* ef = enc_feat + ((size_t)b * L_ + l) * H_;
    float s = 0.0f;
#pragma unroll 4
    for (int hh = lane; hh < H_; hh += 32)
      s += s_v[hh] * tanhf(ef[hh] + s_dec[hh]);
#pragma unroll
    for (int off = 16; off; off >>= 1) s += __shfl_xor(s, off, 32);
    if (lane == 0) s_e[l] = s;
  }
  __syncthreads();

  // ---- masked, renormalized softmax (softmax Z cancels in renorm) ----
  float lm = -INFINITY;
  for (int i = tid; i < L_; i += 1024) lm = fmaxf(lm, s_e[i]);
#pragma unroll
  for (int off = 16; off; off >>= 1) lm = fmaxf(lm, __shfl_xor(lm, off, 32));
  if (lane == 0) s_red[wave] = lm;
  __syncthreads();
  if (wave == 0) {
    float m = s_red[lane];
#pragma unroll
    for (int off = 16; off; off >>= 1) m = fmaxf(m, __shfl_xor(m, off, 32));
    if (lane == 0) s_bcast = m;
  }
  __syncthreads();
  const float gmax = s_bcast;

  float ps = 0.0f;
  for (int i = tid; i < L_; i += 1024) {
    const float w = __expf(s_e[i] - gmax) * mask[(size_t)b * L_ + i];
    s_e[i] = w;
    ps += w;
  }
#pragma unroll
  for (int off = 16; off; off >>= 1) ps += __shfl_xor(ps, off, 32);
  __syncthreads();              // done reading s_red from max phase
  if (lane == 0) s_red[wave] = ps;
  __syncthreads();
  if (wave == 0) {
    float s = s_red[lane];
#pragma unroll
    for (int off = 16; off; off >>= 1) s += __shfl_xor(s, off, 32);
    if (lane == 0) s_bcast = s;
  }
  __syncthreads();
  const float inv = 1.0f / s_bcast;

  for (int i = tid; i < L_; i += 1024) {
    const float a = s_e[i] * inv;
    s_e[i] = a;
    attn_out[(size_t)b * L_ + i] = a;
  }
  __syncthreads();

  // ---- ctx[b, j] = sum_l a[l] * enc_states[b, l, j], j in [0, 2H) ----
  // thread j handles column j: per-l reads are fully coalesced (4KB lines).
  float acc = 0.0f;
  const float* es = enc_states + (size_t)b * L_ * 2 * H_ + tid;
#pragma unroll 4
  for (int l = 0; l < L_; ++l) acc += s_e[l] * es[(size_t)l * 2 * H_];
  hctx[(size_t)b * 3 * H_ + H_ + tid] = acc;
}

// ---------------------------------------------------------------------------
// p_gen = sigmoid([ctx, h, c, x] . Wpg + b) — one wave per batch row.
// ---------------------------------------------------------------------------
__global__ __launch_bounds__(256) void pgen_kernel(
    const float* __restrict__ hctx,   // [B,3H]: h at 0, ctx at H
    const float* __restrict__ c,      // [B,H]
    const float* __restrict__ x,      // [B,E] (x_all at step t)
    const float* __restrict__ Wpg, const float* __restrict__ Wpg_b,
    float* __restrict__ p_out)        // [B] slice at step t
{
  const int lane = threadIdx.x & 31;
  const int b = (blockIdx.x * blockDim.x + threadIdx.x) >> 5;
  const float* hrow = hctx + (size_t)b * 3 * H_;
  float s = 0.0f;
  for (int k = lane; k < 2 * H_; k += 32) s += hrow[H_ + k] * Wpg[k];               // ctx
  for (int k = lane; k < H_; k += 32)     s += hrow[k] * Wpg[2 * H_ + k];           // h
  for (int k = lane; k < H_; k += 32)     s += c[(size_t)b * H_ + k] * Wpg[3 * H_ + k]; // c
  for (int k = lane; k < E_; k += 32)     s += x[(size_t)b * E_ + k] * Wpg[4 * H_ + k]; // x
#pragma unroll
  for (int off = 16; off; off >>= 1) s += __shfl_xor(s, off, 32);
  if (lane == 0) p_out[b] = sigm(s + Wpg_b[0]);
}

// ---------------------------------------------------------------------------
extern "C" void kernel_launch(void* const* d_in, const int* in_sizes, int n_in,
                              void* d_out, int out_size, void* d_ws, size_t ws_size,
                              hipStream_t stream) {
  const float* dec_inputs = (const float*)d_in[0];   // [T,B,E]
  const float* initial_h  = (const float*)d_in[1];   // [B,H]
  const float* initial_c  = (const float*)d_in[2];   // [B,H]
  const float* enc_states = (const float*)d_in[3];   // [B,L,2H]
  const float* pad_mask   = (const float*)d_in[4];   // [B,L]
  const float* Ws_w  = (const float*)d_in[5];        // [H,2H]
  const float* Ws_b  = (const float*)d_in[6];
  const float* in2x_w = (const float*)d_in[7];       // [E,2H+E], only [:, :E] matters
  const float* in2x_b = (const float*)d_in[8];
  const float* Wpg_w = (const float*)d_in[9];        // [1,4H+E]
  const float* Wpg_b = (const float*)d_in[10];
  const float* Wout_w = (const float*)d_in[11];      // [H,3H]
  const float* Wout_b = (const float*)d_in[12];
  const float* Wh_w = (const float*)d_in[13];        // [H,2H]
  const float* Wh_b = (const float*)d_in[14];
  const float* v_vec = (const float*)d_in[15];       // [H]
  const float* W_ih = (const float*)d_in[16];        // [4H,E]
  const float* W_hh = (const float*)d_in[17];        // [4H,H]
  const float* b_ih = (const float*)d_in[18];
  const float* b_hh = (const float*)d_in[19];

  float* out = (float*)d_out;
  float* out_outputs = out;                                    // [T,B,H]
  float* out_h       = out + (size_t)T_ * B_ * H_;             // [B,H]
  float* out_c       = out_h + (size_t)B_ * H_;                // [B,H]
  float* out_attn    = out_c + (size_t)B_ * H_;                // [T,B,L]
  float* out_pgen    = out_attn + (size_t)T_ * B_ * L_;        // [T,B]

  // workspace carve-up (~58 MB)
  float* ws = (float*)d_ws;
  float* enc_feat = ws;                               ws += (size_t)B_ * L_ * H_;      // 13.1M
  float* x_all    = ws;                               ws += (size_t)T_ * B_ * E_;      // 1.05M
  float* gates    = ws;                               ws += (size_t)B_ * 4 * H_;
  float* h_buf    = ws;                               ws += (size_t)B_ * H_;
  float* c_buf    = ws;                               ws += (size_t)B_ * H_;
  float* hc_buf   = ws;                               ws += (size_t)B_ * 2 * H_;
  float* hctx_buf = ws;                               ws += (size_t)B_ * 3 * H_;
  float* dec_buf  = ws;                               ws += (size_t)B_ * H_;

  // ---- init recurrent state (don't mutate inputs) ----
  hipMemcpyAsync(h_buf, initial_h, (size_t)B_ * H_ * sizeof(float),
                 hipMemcpyDeviceToDevice, stream);
  hipMemcpyAsync(c_buf, initial_c, (size_t)B_ * H_ * sizeof(float),
                 hipMemcpyDeviceToDevice, stream);

  // ---- precompute enc_feat = enc_states @ Wh_w^T + Wh_b : M=25600,N=512,K=1024
  wmma_gemm_f32<<<dim3(H_ / 16, (B_ * L_) / 64), 128, 0, stream>>>(
      enc_states, 2 * H_, Wh_w, 2 * H_, 2 * H_,
      nullptr, 0, nullptr, 0, 0, Wh_b, enc_feat, H_);

  // ---- precompute x_all = dec_inputs @ in2x_w[:, :E]^T + b : M=4096,N=256,K=256
  wmma_gemm_f32<<<dim3(E_ / 16, (T_ * B_) / 64), 128, 0, stream>>>(
      dec_inputs, E_, in2x_w, 2 * H_ + E_, E_,
      nullptr, 0, nullptr, 0, 0, in2x_b, x_all, E_);

  // ---- sequential decode ----
  for (int t = 0; t < T_; ++t) {
    const float* x_t = x_all + (size_t)t * B_ * E_;

    // gates = x_t @ W_ih^T + h @ W_hh^T   (biases added in lstm_cell)
    wmma_gemm_f32<<<dim3((4 * H_) / 16, B_ / 64), 128, 0, stream>>>(
        x_t, E_, W_ih, E_, E_,
        h_buf, H_, W_hh, H_, H_, nullptr, gates, 4 * H_);

    lstm_cell<<<(B_ * H_) / 256, 256, 0, stream>>>(
        gates, b_ih, b_hh, h_buf, c_buf, hc_buf, hctx_buf);

    // dec = [h,c] @ Ws_w^T + Ws_b : M=64,N=512,K=1024
    wmma_gemm_f32<<<dim3(H_ / 16, B_ / 64), 128, 0, stream>>>(
        hc_buf, 2 * H_, Ws_w, 2 * H_, 2 * H_,
        nullptr, 0, nullptr, 0, 0, Ws_b, dec_buf, H_);

    // attention: scores + masked renorm softmax + ctx (into hctx[:, H:3H])
    attn_fused<<<B_, 1024, 0, stream>>>(
        enc_feat, dec_buf, v_vec, pad_mask, enc_states,
        out_attn + (size_t)t * B_ * L_, hctx_buf);

    // out_t = [h, ctx] @ Wout_w^T + Wout_b : M=64,N=512,K=1536 -> d_out
    wmma_gemm_f32<<<dim3(H_ / 16, B_ / 64), 128, 0, stream>>>(
        hctx_buf, 3 * H_, Wout_w, 3 * H_, 3 * H_,
        nullptr, 0, nullptr, 0, 0, Wout_b,
        out_outputs + (size_t)t * B_ * H_, H_);

    // p_gen
    pgen_kernel<<<(B_ * 32) / 256, 256, 0, stream>>>(
        hctx_buf, c_buf, x_t, Wpg_w, Wpg_b, out_pgen + (size_t)t * B_);
  }

  // ---- final h, c ----
  hipMemcpyAsync(out_h, h_buf, (size_t)B_ * H_ * sizeof(float),
                 hipMemcpyDeviceToDevice, stream);
  hipMemcpyAsync(out_c, c_buf, (size_t)B_ * H_ * sizeof(float),
                 hipMemcpyDeviceToDevice, stream);
}